// RNN_84009560310330
// MI455X (gfx1250) — compile-verified
//
#include <hip/hip_runtime.h>
#include <hip/hip_bf16.h>

// Problem constants (from the reference): B=64, T=32768, D_IN=7, H=10, D_OUT=1, L=2
#define B_SZ 64
#define T_SZ 32768
#define DIN  7
#define HSZ  10

typedef __attribute__((ext_vector_type(2))) float v2f;
typedef __attribute__((ext_vector_type(8))) float v8f;

// Hidden-index relabeling that makes the WMMA C/D accumulator layout directly
// reusable as the B operand of the next 16x16x4 f32 WMMA (C VGPR i -> B slot
// (chunk i>>1, v i&1)), with an *identity* column map for logical k = 0..9:
//   hardware C-row r holds logical hidden index rowmap[r] (-1 = zero row).
__device__ __constant__ const int c_rowmap[16] = {0, 1, 4, 5, 8,  9,  -1, -1,
                                                  2, 3, 6, 7, -1, -1, -1, -1};

// ---- lane-permute: swap lower/upper half-wave (ds_swizzle SWAPX16 = 0x401f) ----
__device__ __forceinline__ float swapx16(float v) {
  return __builtin_bit_cast(
      float, __builtin_amdgcn_ds_swizzle(__builtin_bit_cast(int, v), 0x401f));
}

// ---- hardware tanh (gfx1250 V_TANH_F32) ----
__device__ __forceinline__ float tanh_hw(float x) {
#if __has_builtin(__builtin_amdgcn_tanhf)
  return __builtin_amdgcn_tanhf(x);
#else
  float r;
  asm("v_tanh_f32 %0, %1\n\tv_nop" : "=v"(r) : "v"(x));
  return r;
#endif
}

// ---- D = A(16x4) * B(4x16) + C, f32 ----
__device__ __forceinline__ v8f wmma4(v2f a, v2f b, v8f c) {
  return __builtin_amdgcn_wmma_f32_16x16x4_f32(false, a, false, b, (short)0, c,
                                               false, false);
}

// Weight element for A operand: logical row lrow (already row-permuted),
// hardware-k position kpos (identity column map; OOB -> 0).
__device__ __forceinline__ float wval(const float* __restrict__ W, int lrow,
                                      int kpos, int cols) {
  return (lrow >= 0 && kpos < cols) ? W[lrow * cols + kpos] : 0.f;
}

// A-operand K-chunk c: lane holds hardware row j = lane&15,
// v0 -> kpos = 4c + 2*(lane>>4), v1 -> kpos+1.
__device__ __forceinline__ v2f load_a(const float* __restrict__ W, int lrow,
                                      int cols, int c, int hi) {
  int kp = 4 * c + 2 * hi;
  v2f a;
  a.x = wval(W, lrow, kp, cols);
  a.y = wval(W, lrow, kp + 1, cols);
  return a;
}

__global__ __launch_bounds__(32) void rnn_scan_wmma(
    const float* __restrict__ x, const float* __restrict__ W_ih0,
    const float* __restrict__ W_hh0, const float* __restrict__ b_ih0,
    const float* __restrict__ b_hh0, const float* __restrict__ W_ih1,
    const float* __restrict__ W_hh1, const float* __restrict__ b_ih1,
    const float* __restrict__ b_hh1, const float* __restrict__ W_fc,
    const float* __restrict__ b_fc, float* __restrict__ out) {
  const int lane = threadIdx.x & 31;
  const int lo   = lane & 15;
  const int hi   = lane >> 4;
  const int bt   = blockIdx.x;      // batch tile 0..3 (one wave each)
  const int b    = bt * 16 + lo;    // this lane-column's batch element

  const int lrow = c_rowmap[lo];    // logical hidden row for A hardware row lo

  // ---- constant A-operands (row-permuted), resident in VGPRs for the scan ----
  v2f Aih0[2], Ahh0[3], Aih1[3], Ahh1[3];
  for (int c = 0; c < 2; ++c) Aih0[c] = load_a(W_ih0, lrow, DIN, c, hi);
  for (int c = 0; c < 3; ++c) {
    Ahh0[c] = load_a(W_hh0, lrow, HSZ, c, hi);
    Aih1[c] = load_a(W_ih1, lrow, HSZ, c, hi);
    Ahh1[c] = load_a(W_hh1, lrow, HSZ, c, hi);
  }
  v8f bias0, bias1;
  for (int i = 0; i < 8; ++i) {
    int l    = c_rowmap[i + 8 * hi];      // logical row in C VGPR i, this half
    bias0[i] = (l >= 0) ? (b_ih0[l] + b_hh0[l]) : 0.f;
    bias1[i] = (l >= 0) ? (b_ih1[l] + b_hh1[l]) : 0.f;
  }

  // ---- per-lane x addressing (identity d at hardware-k; k=7 column is 0) ----
  const float* xb = x + (size_t)b * ((size_t)T_SZ * DIN);
  const int d0 = hi ? 2 : 0;
  const int d1 = hi ? 3 : 1;
  const int d2 = hi ? 6 : 4;
  const int d3 = 5;                 // hi-half k=7 slot: reuse d=5, weight is 0

  v8f H0 = {};  // layer-0 hidden state, permuted C/D layout == B layout
  v8f H1 = {};  // layer-1 hidden state
  const v8f Z = {};

  // software pipeline: preload x row for t=0
  float xv0 = xb[d0], xv1 = xb[d1], xv2 = xb[d2], xv3 = xb[d3];

  for (int t = 0; t < T_SZ; ++t) {
    // pull this batch-row's stream ~16 steps ahead (global_prefetch_b8)
    if (t + 16 < T_SZ)
      __builtin_prefetch(&xb[(size_t)(t + 16) * DIN], 0, 1);

    v2f Bx0 = {xv0, xv1};  // hardware k = 0/2, 1/3
    v2f Bx1 = {xv2, xv3};  // hardware k = 4/6, 5/(7 -> zero weight)

    // preload next step's x row (off the critical path)
    if (t + 1 < T_SZ) {
      size_t o = (size_t)(t + 1) * DIN;
      xv0 = xb[o + d0]; xv1 = xb[o + d1];
      xv2 = xb[o + d2]; xv3 = xb[o + d3];
    }

    // H state VGPRs serve directly as B-operand K-chunks (no permutes).
    v2f h0c0 = {H0[0], H0[1]}, h0c1 = {H0[2], H0[3]}, h0c2 = {H0[4], H0[5]};

    // ---- layer 0: bias + W_ih0*x_t^T (early chain) + W_hh0*H0 (3 parallel) --
    v8f aP = wmma4(Aih0[0], Bx0, bias0);
    aP     = wmma4(Aih0[1], Bx1, aP);
    v8f aA = wmma4(Ahh0[0], h0c0, aP);   // only 1 H-dependent WMMA per chain
    v8f aB = wmma4(Ahh0[1], h0c1, Z);
    v8f aC = wmma4(Ahh0[2], h0c2, Z);
    for (int i = 0; i < 6; ++i) H0[i] = tanh_hw((aA[i] + aB[i]) + aC[i]);
    H0[6] = 0.f;  // C rows with invalid logical index are exactly 0
    H0[7] = 0.f;

    v2f g0 = {H0[0], H0[1]}, g1 = {H0[2], H0[3]}, g2 = {H0[4], H0[5]};
    v2f h1c0 = {H1[0], H1[1]}, h1c1 = {H1[2], H1[3]}, h1c2 = {H1[4], H1[5]};

    // ---- layer 1: bias + W_ih1*H0_new (feed-forward chain) + W_hh1*H1 ------
    v8f bP = wmma4(Aih1[0], g0, bias1);
    bP     = wmma4(Aih1[1], g1, bP);
    bP     = wmma4(Aih1[2], g2, bP);
    v8f bA = wmma4(Ahh1[0], h1c0, bP);   // only 1 H1-dependent WMMA per chain
    v8f bB = wmma4(Ahh1[1], h1c1, Z);
    v8f bC = wmma4(Ahh1[2], h1c2, Z);
    for (int i = 0; i < 6; ++i) H1[i] = tanh_hw((bA[i] + bB[i]) + bC[i]);
    H1[6] = 0.f;
    H1[7] = 0.f;
  }

  // ---- final FC: out[b] = sum_j W_fc[j] * H1[j, b] + b_fc ----
  float s = 0.f;
  for (int i = 0; i < 8; ++i) {
    int l   = c_rowmap[i + 8 * hi];
    float w = (l >= 0) ? W_fc[l] : 0.f;
    s += w * H1[i];
  }
  s += swapx16(s);  // combine the two half-wave partial sums
  if (hi == 0) out[b] = s + b_fc[0];
}

extern "C" void kernel_launch(void* const* d_in, const int* in_sizes, int n_in,
                              void* d_out, int out_size, void* d_ws,
                              size_t ws_size, hipStream_t stream) {
  const float* x     = (const float*)d_in[0];
  const float* W_ih0 = (const float*)d_in[1];
  const float* W_hh0 = (const float*)d_in[2];
  const float* b_ih0 = (const float*)d_in[3];
  const float* b_hh0 = (const float*)d_in[4];
  const float* W_ih1 = (const float*)d_in[5];
  const float* W_hh1 = (const float*)d_in[6];
  const float* b_ih1 = (const float*)d_in[7];
  const float* b_hh1 = (const float*)d_in[8];
  const float* W_fc  = (const float*)d_in[9];
  const float* b_fc  = (const float*)d_in[10];
  float* out = (float*)d_out;

  // 4 batch tiles of 16 -> 4 independent single-wave workgroups; the scan is
  // latency-bound on the sequential T chain, so each wave runs its own tile.
  rnn_scan_wmma<<<dim3(4), dim3(32), 0, stream>>>(
      x, W_ih0, W_hh0, b_ih0, b_hh0, W_ih1, W_hh1, b_ih1, b_hh1, W_fc, b_fc,
      out);
}